// RGCN_10720238370917
// MI455X (gfx1250) — compile-verified
//
#include <hip/hip_runtime.h>
#include <hip/hip_bf16.h>
#include <math.h>

typedef __attribute__((ext_vector_type(2))) float v2f;
typedef __attribute__((ext_vector_type(8))) float v8f;

#define RELS 90

// ---------------------------------------------------------------- utilities
__global__ void zero_words_kernel(unsigned* __restrict__ p, size_t n) {
    size_t i = (size_t)blockIdx.x * blockDim.x + threadIdx.x;
    size_t stride = (size_t)gridDim.x * blockDim.x;
    for (; i < n; i += stride) p[i] = 0u;
}

// count edges per (dst, relation) segment for 'mean' normalization
__global__ __launch_bounds__(256) void count_kernel(
    const int* __restrict__ ei, const int* __restrict__ et,
    unsigned* __restrict__ cnt, int E) {
    int e = blockIdx.x * blockDim.x + threadIdx.x;
    if (e >= E) return;
    int dst = ei[E + e];
    int t   = et[e];
    atomicAdd(&cnt[(size_t)dst * RELS + t], 1u);
}

// convert counts in place to 1/max(cnt,1) so edge kernels avoid fp divides
__global__ void invcnt_kernel(unsigned* __restrict__ p, size_t n) {
    size_t i = (size_t)blockIdx.x * blockDim.x + threadIdx.x;
    size_t stride = (size_t)gridDim.x * blockDim.x;
    for (; i < n; i += stride) {
        float c = (float)p[i];
        ((float*)p)[i] = 1.0f / fmaxf(c, 1.0f);
    }
}

// ---------------------------------------------------------------- edge kernels
// Layer 1: in=4 (B=2,ib=2,ob=4) -> out=8, mean aggregation
__global__ __launch_bounds__(256) void edge1_kernel(
    const float* __restrict__ x, const int* __restrict__ ei,
    const int* __restrict__ et, const float* __restrict__ w,
    const float* __restrict__ inv, float* __restrict__ h1, int E) {
    __shared__ float Ws[RELS * 16];
    for (int i = threadIdx.x; i < RELS * 16; i += blockDim.x) Ws[i] = w[i];
    __syncthreads();
    int e = blockIdx.x * blockDim.x + threadIdx.x;
    if (e >= E) return;
    int src = ei[e], dst = ei[E + e], t = et[e];
    float4 xv = *(const float4*)(x + (size_t)src * 4);
    const float* W = &Ws[t * 16];
    float scale = inv[(size_t)dst * RELS + t];
    float* hb = h1 + (size_t)dst * 8;
#pragma unroll
    for (int o = 0; o < 4; ++o) {
        atomicAdd(&hb[o],     (xv.x * W[o]      + xv.y * W[4 + o])  * scale);
        atomicAdd(&hb[4 + o], (xv.z * W[8 + o]  + xv.w * W[12 + o]) * scale);
    }
}

// Layer 2: in=8 (B=4,ib=2,ob=3) -> out=12, add aggregation
__global__ __launch_bounds__(256) void edge2_kernel(
    const float* __restrict__ h1, const int* __restrict__ ei,
    const int* __restrict__ et, const float* __restrict__ w,
    float* __restrict__ h2, int E) {
    __shared__ float Ws[RELS * 24];
    for (int i = threadIdx.x; i < RELS * 24; i += blockDim.x) Ws[i] = w[i];
    __syncthreads();
    int e = blockIdx.x * blockDim.x + threadIdx.x;
    if (e >= E) return;
    int src = ei[e], dst = ei[E + e], t = et[e];
    const float4* hp = (const float4*)(h1 + (size_t)src * 8);
    float4 v0 = hp[0], v1 = hp[1];
    float xi[8] = {v0.x, v0.y, v0.z, v0.w, v1.x, v1.y, v1.z, v1.w};
    const float* W = &Ws[t * 24];
    float* hb = h2 + (size_t)dst * 12;
#pragma unroll
    for (int b = 0; b < 4; ++b) {
        float a0 = xi[2 * b], a1 = xi[2 * b + 1];
#pragma unroll
        for (int o = 0; o < 3; ++o) {
            float m = a0 * W[(2 * b) * 3 + o] + a1 * W[(2 * b + 1) * 3 + o];
            atomicAdd(&hb[3 * b + o], m);
        }
    }
}

// Layer 3: in=12 (B=2,ib=6,ob=2) -> out=4, mean aggregation
__global__ __launch_bounds__(256) void edge3_kernel(
    const float* __restrict__ h2, const int* __restrict__ ei,
    const int* __restrict__ et, const float* __restrict__ w,
    const float* __restrict__ inv, float* __restrict__ out, int E) {
    __shared__ float Ws[RELS * 24];
    for (int i = threadIdx.x; i < RELS * 24; i += blockDim.x) Ws[i] = w[i];
    __syncthreads();
    int e = blockIdx.x * blockDim.x + threadIdx.x;
    if (e >= E) return;
    int src = ei[e], dst = ei[E + e], t = et[e];
    const float4* hp = (const float4*)(h2 + (size_t)src * 12);
    float4 v0 = hp[0], v1 = hp[1], v2 = hp[2];
    float xi[12] = {v0.x, v0.y, v0.z, v0.w, v1.x, v1.y, v1.z, v1.w,
                    v2.x, v2.y, v2.z, v2.w};
    const float* W = &Ws[t * 24];
    float scale = inv[(size_t)dst * RELS + t];
    float* ob = out + (size_t)dst * 4;
#pragma unroll
    for (int b = 0; b < 2; ++b) {
#pragma unroll
        for (int o = 0; o < 2; ++o) {
            float m = 0.0f;
#pragma unroll
            for (int i = 0; i < 6; ++i)
                m += xi[6 * b + i] * W[(b * 6 + i) * 2 + o];
            atomicAdd(&ob[2 * b + o], m * scale);
        }
    }
}

// ---------------------------------------------------------------- WMMA root kernels
// V_WMMA_F32_16X16X4_F32 layouts (wave32):
//   A 16x4:  lanes 0-15 hold K={0,1} (vgpr0,1), lanes 16-31 hold K={2,3}
//   B 4x16:  vgpr0 = rows K=0 (lanes 0-15) / K=2 (lanes 16-31); vgpr1 = K=1 / K=3
//   C/D 16x16: vgpr v, lanes 0-15 -> row v, lanes 16-31 -> row 8+v
// Padded columns (n >= OUT) carry clamped duplicates: D[:,n] depends only on
// B[:,n], C[:,n], so valid columns are unaffected; stores are predicated.

// h1[n,0:8] = relu(h1_acc[n] + x[n,0:4] @ root(4x8) + bias)   [full 16-row tiles]
__global__ __launch_bounds__(256) void root1_kernel(
    const float* __restrict__ x, const float* __restrict__ root,
    const float* __restrict__ bias, float* __restrict__ h1, int numTiles) {
    int wave = threadIdx.x >> 5, lane = threadIdx.x & 31;
    int tile = blockIdx.x * (blockDim.x >> 5) + wave;
    if (tile >= numTiles) return;
    int half = lane >> 4, n = lane & 15;
    int nc = n < 8 ? n : 7;
    int base = tile * 16;

    v2f a = *(const v2f*)(x + (size_t)(base + n) * 4 + 2 * half);
    v2f b;
    b.x = root[(2 * half + 0) * 8 + nc];
    b.y = root[(2 * half + 1) * 8 + nc];
    float bb = bias[nc];
    float* cp = h1 + (size_t)(base + half * 8) * 8;
    v8f c;
#pragma unroll
    for (int v = 0; v < 8; ++v) c[v] = cp[v * 8 + nc];
    v8f d = __builtin_amdgcn_wmma_f32_16x16x4_f32(false, a, false, b,
                                                  (short)0, c, false, false);
    if (n < 8) {
#pragma unroll
        for (int v = 0; v < 8; ++v) cp[v * 8 + n] = fmaxf(d[v] + bb, 0.0f);
    }
}

// h2[n,0:12] = relu(h2_acc[n] + h1[n,0:8] @ root(8x12) + bias) : two chained WMMAs
__global__ __launch_bounds__(256) void root2_kernel(
    const float* __restrict__ h1, const float* __restrict__ root,
    const float* __restrict__ bias, float* __restrict__ h2, int numTiles) {
    int wave = threadIdx.x >> 5, lane = threadIdx.x & 31;
    int tile = blockIdx.x * (blockDim.x >> 5) + wave;
    if (tile >= numTiles) return;
    int half = lane >> 4, n = lane & 15;
    int nc = n < 12 ? n : 11;
    int base = tile * 16;

    const float* ap = h1 + (size_t)(base + n) * 8 + 2 * half;
    v2f a0 = *(const v2f*)(ap);
    v2f a1 = *(const v2f*)(ap + 4);
    v2f b0, b1;
    b0.x = root[(2 * half + 0) * 12 + nc];
    b0.y = root[(2 * half + 1) * 12 + nc];
    b1.x = root[(4 + 2 * half + 0) * 12 + nc];
    b1.y = root[(4 + 2 * half + 1) * 12 + nc];
    float bb = bias[nc];
    float* cp = h2 + (size_t)(base + half * 8) * 12;
    v8f c;
#pragma unroll
    for (int v = 0; v < 8; ++v) c[v] = cp[v * 12 + nc];
    v8f d = __builtin_amdgcn_wmma_f32_16x16x4_f32(false, a0, false, b0,
                                                  (short)0, c, false, false);
    d = __builtin_amdgcn_wmma_f32_16x16x4_f32(false, a1, false, b1,
                                              (short)0, d, false, false);
    if (n < 12) {
#pragma unroll
        for (int v = 0; v < 8; ++v) cp[v * 12 + n] = fmaxf(d[v] + bb, 0.0f);
    }
}

// out[n,0:4] = out_acc[n] + h2[n,0:12] @ root(12x4) + bias : three chained WMMAs
__global__ __launch_bounds__(256) void root3_kernel(
    const float* __restrict__ h2, const float* __restrict__ root,
    const float* __restrict__ bias, float* __restrict__ out, int numTiles) {
    int wave = threadIdx.x >> 5, lane = threadIdx.x & 31;
    int tile = blockIdx.x * (blockDim.x >> 5) + wave;
    if (tile >= numTiles) return;
    int half = lane >> 4, n = lane & 15;
    int nc = n < 4 ? n : 3;
    int base = tile * 16;

    const float* ap = h2 + (size_t)(base + n) * 12 + 2 * half;
    v2f a[3], b[3];
#pragma unroll
    for (int j = 0; j < 3; ++j) {
        int k = j * 4 + 2 * half;
        a[j] = *(const v2f*)(ap + j * 4);
        b[j].x = root[(k + 0) * 4 + nc];
        b[j].y = root[(k + 1) * 4 + nc];
    }
    float bb = bias[nc];
    float* cp = out + (size_t)(base + half * 8) * 4;
    v8f c;
#pragma unroll
    for (int v = 0; v < 8; ++v) c[v] = cp[v * 4 + nc];
    v8f d = __builtin_amdgcn_wmma_f32_16x16x4_f32(false, a[0], false, b[0],
                                                  (short)0, c, false, false);
    d = __builtin_amdgcn_wmma_f32_16x16x4_f32(false, a[1], false, b[1],
                                              (short)0, d, false, false);
    d = __builtin_amdgcn_wmma_f32_16x16x4_f32(false, a[2], false, b[2],
                                              (short)0, d, false, false);
    if (n < 4) {
#pragma unroll
        for (int v = 0; v < 8; ++v) cp[v * 4 + n] = d[v] + bb;  // logits
    }
}

// scalar tail for nodes beyond the last full 16-row tile (not launched when N%16==0)
__global__ void root_tail_kernel(const float* __restrict__ xin,
                                 const float* __restrict__ root,
                                 const float* __restrict__ bias,
                                 float* __restrict__ acc,
                                 int IN, int OUT, int start, int N, int doRelu) {
    int i = start + blockIdx.x * blockDim.x + threadIdx.x;
    if (i >= N) return;
    for (int o = 0; o < OUT; ++o) {
        float s = acc[(size_t)i * OUT + o] + bias[o];
        for (int k = 0; k < IN; ++k) s += xin[(size_t)i * IN + k] * root[k * OUT + o];
        acc[(size_t)i * OUT + o] = doRelu ? fmaxf(s, 0.0f) : s;
    }
}

// in-place log_softmax over the 4 classes of each node
__global__ __launch_bounds__(256) void logsoftmax_kernel(float* __restrict__ out, int N) {
    int i = blockIdx.x * blockDim.x + threadIdx.x;
    if (i >= N) return;
    float4 v = ((float4*)out)[i];
    float m = fmaxf(fmaxf(v.x, v.y), fmaxf(v.z, v.w));
    float s = expf(v.x - m) + expf(v.y - m) + expf(v.z - m) + expf(v.w - m);
    float l = m + logf(s);
    ((float4*)out)[i] = make_float4(v.x - l, v.y - l, v.z - l, v.w - l);
}

// ---------------------------------------------------------------- launcher
extern "C" void kernel_launch(void* const* d_in, const int* in_sizes, int n_in,
                              void* d_out, int out_size, void* d_ws, size_t ws_size,
                              hipStream_t stream) {
    const float* x     = (const float*)d_in[0];
    const int*   ei    = (const int*)d_in[1];
    const int*   et    = (const int*)d_in[2];
    const float* w1    = (const float*)d_in[3];
    const float* root1 = (const float*)d_in[4];
    const float* b1    = (const float*)d_in[5];
    const float* w2    = (const float*)d_in[6];
    const float* root2 = (const float*)d_in[7];
    const float* b2    = (const float*)d_in[8];
    const float* w3    = (const float*)d_in[9];
    const float* root3 = (const float*)d_in[10];
    const float* b3    = (const float*)d_in[11];

    const int N = in_sizes[0] / 4;   // 100000
    const int E = in_sizes[1] / 2;   // 3200000
    float* out = (float*)d_out;

    // workspace: counts/inv [N*R] u32->f32 | h1 [N*8] f32 | h2 [N*12] f32  (~44 MB)
    unsigned* cnt = (unsigned*)d_ws;
    float* inv = (float*)d_ws;
    float* h1 = (float*)((char*)d_ws + (size_t)N * RELS * sizeof(unsigned));
    float* h2 = h1 + (size_t)N * 8;

    const size_t segWords = (size_t)N * RELS;
    const size_t wsWords = segWords + (size_t)N * 8 + (size_t)N * 12;
    zero_words_kernel<<<4096, 256, 0, stream>>>((unsigned*)d_ws, wsWords);
    zero_words_kernel<<<512, 256, 0, stream>>>((unsigned*)d_out, (size_t)N * 4);

    const int eBlocks = (E + 255) / 256;
    count_kernel<<<eBlocks, 256, 0, stream>>>(ei, et, cnt, E);
    invcnt_kernel<<<4096, 256, 0, stream>>>(cnt, segWords);

    const int fullTiles = N / 16;             // 6250 (exact for N=100000)
    const int tail = N - fullTiles * 16;      // 0 here
    const int tBlocks = (fullTiles + 7) / 8;  // 8 waves per 256-thread block

    edge1_kernel<<<eBlocks, 256, 0, stream>>>(x, ei, et, w1, inv, h1, E);
    root1_kernel<<<tBlocks, 256, 0, stream>>>(x, root1, b1, h1, fullTiles);
    if (tail) root_tail_kernel<<<1, 32, 0, stream>>>(x, root1, b1, h1, 4, 8,
                                                     fullTiles * 16, N, 1);

    edge2_kernel<<<eBlocks, 256, 0, stream>>>(h1, ei, et, w2, h2, E);
    root2_kernel<<<tBlocks, 256, 0, stream>>>(h1, root2, b2, h2, fullTiles);
    if (tail) root_tail_kernel<<<1, 32, 0, stream>>>(h1, root2, b2, h2, 8, 12,
                                                     fullTiles * 16, N, 1);

    edge3_kernel<<<eBlocks, 256, 0, stream>>>(h2, ei, et, w3, inv, out, E);
    root3_kernel<<<tBlocks, 256, 0, stream>>>(h2, root3, b3, out, fullTiles);
    if (tail) root_tail_kernel<<<1, 32, 0, stream>>>(h2, root3, b3, out, 12, 4,
                                                     fullTiles * 16, N, 0);

    logsoftmax_kernel<<<(N + 255) / 256, 256, 0, stream>>>(out, N);
}